// QuantumDiscriminator_44581760532619
// MI455X (gfx1250) — compile-verified
//
#include <hip/hip_runtime.h>
#include <math.h>

// ---------------------------------------------------------------------------
// QuantumDiscriminator fused kernel for MI455X (gfx1250, wave32, WMMA)
//   out[s] = sigmoid( (x^T Q x) / ||x||^2 ),  x = tanh(inp[s] @ W + b)
// Phase 1: 16x1024 tile of tanh(GEMM) via V_WMMA_F32_16X16X4_F32 into LDS.
// Phase 2: 30 unitary 2x2 butterflies (regs + wave32 shuffles) + CZ diagonals
//          + quadratic expectation readout, one wave per row, state in VGPRs.
// ---------------------------------------------------------------------------

typedef __attribute__((ext_vector_type(2))) float v2f;
typedef __attribute__((ext_vector_type(8))) float v8f;

#define SEQ       16384
#define INP_DIM   640
#define QDIM      1024
#define NQ        10
#define NBLOCKS   3

#define ROWS_PER_WG   16
#define WAVES          4
#define THREADS      (WAVES * 32)
#define COLS_PER_WAVE (QDIM / WAVES)          // 256
#define CTILES        (COLS_PER_WAVE / 16)    // 16 col-tiles per wave
#define KSTEPS        (INP_DIM / 4)           // 160 WMMA k-steps

// Branchless tanh with exact saturation: tanh(x) = 1 - 2/(exp(2x)+1).
// __expf -> v_exp_f32 (handles overflow to +inf -> result 1; underflow -> -1),
// rcp -> v_rcp_f32. No EXEC-mask branching, unlike libm tanhf.
__device__ __forceinline__ float tanh_fast(float x) {
  float e = __expf(2.0f * x);
  return 1.0f - 2.0f * __builtin_amdgcn_rcpf(e + 1.0f);
}

// ---- single-qubit gate: G = RZ(ph)*RX(th) applied on wire K --------------
// wire K == bit (9-K) of the amplitude index (MSB wire ordering).
// State layout per wave: amplitude index = r*32 + lane, r = 0..31.
//   bit >= 5  -> pair lives in registers (r, r ^ (1<<(bit-5)))
//   bit <  5  -> pair lives across lanes  (lane ^ (1<<bit)) -> shfl_xor
template <int K>
__device__ __forceinline__ void apply_gate(float (&sr)[32], float (&si)[32],
                                           int lane,
                                           float g00r, float g00i,
                                           float g01r, float g01i,
                                           float g10r, float g10i,
                                           float g11r, float g11i) {
  constexpr int bit = 9 - K;
  if constexpr (bit >= 5) {
    constexpr int R = 1 << (bit - 5);
#pragma unroll
    for (int r = 0; r < 32; ++r) {
      if (!(r & R)) {
        float a0r = sr[r],     a0i = si[r];
        float a1r = sr[r + R], a1i = si[r + R];
        sr[r]     = g00r*a0r - g00i*a0i + g01r*a1r - g01i*a1i;
        si[r]     = g00r*a0i + g00i*a0r + g01r*a1i + g01i*a1r;
        sr[r + R] = g10r*a0r - g10i*a0i + g11r*a1r - g11i*a1i;
        si[r + R] = g10r*a0i + g10i*a0r + g11r*a1i + g11i*a1r;
      }
    }
  } else {
    constexpr int L = 1 << bit;
    const bool hi = (lane & L) != 0;
    const float car = hi ? g11r : g00r, cai = hi ? g11i : g00i; // own-amp coef
    const float cbr = hi ? g10r : g01r, cbi = hi ? g10i : g01i; // partner coef
#pragma unroll
    for (int r = 0; r < 32; ++r) {
      float pr = __shfl_xor(sr[r], L, 32);
      float pi = __shfl_xor(si[r], L, 32);
      float ar = sr[r], ai = si[r];
      sr[r] = car*ar - cai*ai + cbr*pr - cbi*pi;
      si[r] = car*ai + cai*ar + cbr*pi + cbi*pr;
    }
  }
}

__global__ __launch_bounds__(THREADS)
void qdisc_fused_kernel(const float* __restrict__ inp,
                        const float* __restrict__ W,
                        const float* __restrict__ bias,
                        const float* __restrict__ wts,
                        float* __restrict__ out) {
  __shared__ float Y[ROWS_PER_WG][QDIM];  // 64 KB: tanh(GEMM) tile

  const int lane = threadIdx.x & 31;
  const int wave = threadIdx.x >> 5;
  const int row0 = blockIdx.x * ROWS_PER_WG;

  // =====================================================================
  // Phase 1: 16x1024 tile of tanh(inp @ W + b) via V_WMMA_F32_16X16X4_F32
  //   A (16x4): lanes 0-15 hold M=lane, K={0,1} in v[0],v[1];
  //             lanes 16-31 hold M=lane-16, K={2,3}   -> one b64 load
  //   B (4x16): v[0] = B[K0+2*half][N], v[1] = B[K0+1+2*half][N]
  //   D (16x16): d[i] -> row = i + 8*half, col = n0 + (lane&15)
  // =====================================================================
  const int m    = lane & 15;
  const int half = lane >> 4;
  const float* arow = inp + (size_t)(row0 + m) * INP_DIM + 2 * half;

  for (int ct = 0; ct < CTILES; ++ct) {
    const int n = wave * COLS_PER_WAVE + ct * 16 + m;   // output column
    v8f acc = {};
    const float* bcol = W + (size_t)(2 * half) * QDIM + n;
#pragma unroll 8
    for (int kk = 0; kk < KSTEPS; ++kk) {
      v2f a = *(const v2f*)(arow + kk * 4);             // A[m][4kk+2h .. +1]
      v2f bb;
      bb.x = bcol[kk * 4 * QDIM];                       // B[4kk+2h  ][n]
      bb.y = bcol[kk * 4 * QDIM + QDIM];                // B[4kk+2h+1][n]
      acc = __builtin_amdgcn_wmma_f32_16x16x4_f32(
          false, a, false, bb, (short)0, acc, false, false);
    }
    const float bv = bias[n];
#pragma unroll
    for (int i = 0; i < 8; ++i)
      Y[i + 8 * half][n] = tanh_fast(acc[i] + bv);
  }
  __syncthreads();

  // =====================================================================
  // Phase 2: circuit + expectations; each wave owns 4 rows, full state in
  // 64 VGPRs (re/im), amplitude index = r*32 + lane. Gates are linear and
  // expectations quadratic, so normalization is deferred: run on raw x and
  // divide the final scalar by ||x||^2.
  // =====================================================================
  for (int rr = wave; rr < ROWS_PER_WG; rr += WAVES) {
    float sr[32], si[32];
    float nrm = 0.f;
#pragma unroll
    for (int r = 0; r < 32; ++r) {
      float v = Y[rr][r * 32 + lane];
      sr[r] = v; si[r] = 0.f;
      nrm += v * v;                 // ||x||^2 partial
    }

    for (int j = 0; j < NBLOCKS; ++j) {
      const float* wj = wts + j * NQ * 3;
      // 10 single-qubit gates (RZ*RX fused), unrolled over wires
#define DO_GATE(KW)                                                          \
      {                                                                      \
        const float th = wj[(KW) * 3 + 0] * 0.5f;                            \
        const float ph = wj[(KW) * 3 + 1] * 0.5f;                            \
        const float c = cosf(th), s = sinf(th);                              \
        const float cp = cosf(ph), sp = sinf(ph);                            \
        /* g00 = c e^{-iph}; g01 = -i s e^{-iph};                            \
           g10 = -i s e^{iph}; g11 = c e^{iph} */                            \
        apply_gate<KW>(sr, si, lane,                                         \
                       c * cp, -c * sp,                                      \
                       -s * sp, -s * cp,                                     \
                       s * sp, -s * cp,                                      \
                       c * cp, c * sp);                                      \
      }
      DO_GATE(0) DO_GATE(1) DO_GATE(2) DO_GATE(3) DO_GATE(4)
      DO_GATE(5) DO_GATE(6) DO_GATE(7) DO_GATE(8) DO_GATE(9)
#undef DO_GATE
      // CZ ladder (wires k,k+1 for k=0..8): all commute -> one diagonal
      // sign = (-1)^{#adjacent 1-bit pairs in the 10-bit index}
#pragma unroll
      for (int r = 0; r < 32; ++r) {
        int idx = r * 32 + lane;
        if (__popc(idx & (idx >> 1) & 0x1FF) & 1) {
          sr[r] = -sr[r]; si[r] = -si[r];
        }
      }
    }

    // ---- expectations: sum_k <Z_k> + <X_k>, on unnormalized state -------
    float tot = 0.f;
#pragma unroll
    for (int r = 0; r < 32; ++r) {
      int idx = r * 32 + lane;
      float p = sr[r] * sr[r] + si[r] * si[r];
      tot += p * (float)(NQ - 2 * __popc(idx));          // sum_k Z_k
    }
    // X on wires 0..4 (register-stride pairs): factor 2 per unordered pair
#pragma unroll
    for (int kb = 0; kb < 5; ++kb) {
      const int R = 16 >> kb;
#pragma unroll
      for (int r = 0; r < 32; ++r)
        if (!(r & R))
          tot += 2.f * (sr[r] * sr[r + R] + si[r] * si[r + R]);
    }
    // X on wires 5..9 (lane-stride pairs): summing both orientations over
    // all lanes supplies the 2x automatically
#pragma unroll
    for (int kb = 0; kb < 5; ++kb) {
      const int L = 16 >> kb;
#pragma unroll
      for (int r = 0; r < 32; ++r) {
        float pr = __shfl_xor(sr[r], L, 32);
        float pi = __shfl_xor(si[r], L, 32);
        tot += sr[r] * pr + si[r] * pi;
      }
    }

    // wave32 xor-reduction of (tot, nrm)
#pragma unroll
    for (int o = 16; o >= 1; o >>= 1) {
      tot += __shfl_xor(tot, o, 32);
      nrm += __shfl_xor(nrm, o, 32);
    }
    if (lane == 0)
      out[row0 + rr] = 1.f / (1.f + expf(-tot / nrm));
  }
}

extern "C" void kernel_launch(void* const* d_in, const int* in_sizes, int n_in,
                              void* d_out, int out_size, void* d_ws, size_t ws_size,
                              hipStream_t stream) {
  const float* inp = (const float*)d_in[0];   // [16384, 640]
  const float* W   = (const float*)d_in[1];   // [640, 1024]
  const float* b   = (const float*)d_in[2];   // [1024]
  const float* wts = (const float*)d_in[3];   // [3, 10, 3]
  float* out       = (float*)d_out;           // [16384]
  (void)in_sizes; (void)n_in; (void)out_size; (void)d_ws; (void)ws_size;

  dim3 grid(SEQ / ROWS_PER_WG);   // 1024 workgroups
  dim3 block(THREADS);            // 128 threads = 4 waves
  qdisc_fused_kernel<<<grid, block, 0, stream>>>(inp, W, b, wts, out);
}